// SetCriterion_66984309948477
// MI455X (gfx1250) — compile-verified
//
#include <hip/hip_runtime.h>
#include <cstdint>
#include <cstddef>

#define N_COLS 65536
#define NV4    (N_COLS / 4)
#define BLK    1024

__device__ __forceinline__ float frcp(float x) { return __builtin_amdgcn_rcpf(x); }

__device__ __forceinline__ float wred_sum(float v) {
#pragma unroll
  for (int o = 16; o > 0; o >>= 1) v += __shfl_xor(v, o, 32);
  return v;
}
__device__ __forceinline__ float wred_min(float v) {
#pragma unroll
  for (int o = 16; o > 0; o >>= 1) v = fminf(v, __shfl_xor(v, o, 32));
  return v;
}
__device__ __forceinline__ float wred_max(float v) {
#pragma unroll
  for (int o = 16; o > 0; o >>= 1) v = fmaxf(v, __shfl_xor(v, o, 32));
  return v;
}

// 2-sum block reduction (1024 threads = 32 waves), result broadcast to all.
__device__ __forceinline__ void block_red2(float& a, float& b, float* r, int tid) {
  a = wred_sum(a); b = wred_sum(b);
  const int wid = tid >> 5, lane = tid & 31;
  __syncthreads();
  if (lane == 0) { r[wid] = a; r[32 + wid] = b; }
  __syncthreads();
  if (wid == 0) {
    float x = wred_sum(r[lane]);
    float y = wred_sum(r[32 + lane]);
    if (lane == 0) { r[128] = x; r[129] = y; }
  }
  __syncthreads();
  a = r[128]; b = r[129];
}

__device__ __forceinline__ void block_red3(float& a, float& b, float& c, float* r, int tid) {
  a = wred_sum(a); b = wred_sum(b); c = wred_sum(c);
  const int wid = tid >> 5, lane = tid & 31;
  __syncthreads();
  if (lane == 0) { r[wid] = a; r[32 + wid] = b; r[64 + wid] = c; }
  __syncthreads();
  if (wid == 0) {
    float x = wred_sum(r[lane]);
    float y = wred_sum(r[32 + lane]);
    float z = wred_sum(r[64 + lane]);
    if (lane == 0) { r[128] = x; r[129] = y; r[130] = z; }
  }
  __syncthreads();
  a = r[128]; b = r[129]; c = r[130];
}

// sum, sum, min, max
__device__ __forceinline__ void block_red4(float& s0, float& s1, float& mn, float& mx,
                                           float* r, int tid) {
  s0 = wred_sum(s0); s1 = wred_sum(s1); mn = wred_min(mn); mx = wred_max(mx);
  const int wid = tid >> 5, lane = tid & 31;
  __syncthreads();
  if (lane == 0) { r[wid] = s0; r[32 + wid] = s1; r[64 + wid] = mn; r[96 + wid] = mx; }
  __syncthreads();
  if (wid == 0) {
    float a = wred_sum(r[lane]);
    float b = wred_sum(r[32 + lane]);
    float c = wred_min(r[64 + lane]);
    float d = wred_max(r[96 + lane]);
    if (lane == 0) { r[128] = a; r[129] = b; r[130] = c; r[131] = d; }
  }
  __syncthreads();
  s0 = r[128]; s1 = r[129]; mn = r[130]; mx = r[131];
}

// One sweep over the LDS-resident e = exp(-z) row: f = sum(1/(1+e*t)) - K, S = sum(p*(1-p)).
__device__ __forceinline__ void eval_fS(const float4* __restrict__ e4, int tid, float t,
                                        float K, float* r, float& f, float& S) {
  float fs = 0.f, ss = 0.f;
  for (int j = tid; j < NV4; j += BLK) {
    float4 e = e4[j];
    float p0 = frcp(fmaf(e.x, t, 1.f));
    float p1 = frcp(fmaf(e.y, t, 1.f));
    float p2 = frcp(fmaf(e.z, t, 1.f));
    float p3 = frcp(fmaf(e.w, t, 1.f));
    fs += (p0 + p1) + (p2 + p3);
    ss += (fmaf(-p0, p0, p0) + fmaf(-p1, p1, p1)) +
          (fmaf(-p2, p2, p2) + fmaf(-p3, p3, p3));
  }
  block_red2(fs, ss, r, tid);
  f = fs - K;
  S = ss;
}

__device__ __forceinline__ float logit_clipped(float x) {
  x = fminf(fmaxf(x, 1e-6f), 1.f - 1e-6f);
  return __logf(x) - log1pf(-x);
}

__global__ void __launch_bounds__(BLK, 1)
set_criterion_row_kernel(const float* __restrict__ logits,
                         const float* __restrict__ pos,
                         float* __restrict__ ws) {
  extern __shared__ float smem[];   // 65536 floats: z row, then e = exp(-z) in place
  __shared__ float rbuf[136];
  const int tid = threadIdx.x;
  const int row = blockIdx.x;
  const float* zrow = logits + (size_t)row * N_COLS;
  const float* prow = pos + (size_t)row * N_COLS;

  // ---- Async DMA: z row -> LDS (CDNA5 global_load_async_to_lds, ASYNCcnt) ----
#if defined(__HIP_DEVICE_COMPILE__)
  for (int j = tid; j < NV4; j += BLK) {
    uint32_t lds_off = (uint32_t)(size_t)(&smem[4 * j]);  // low 32 bits = LDS offset
    uint32_t goff = (uint32_t)(j * 16);
    asm volatile("global_load_async_to_lds_b128 %0, %1, %2"
                 :: "v"(lds_off), "v"(goff), "s"(zrow) : "memory");
  }
#endif

  // ---- Overlapped with the DMA: K_high = sum(pos row) ----
  float kh = 0.f;
  const float4* p4 = (const float4*)prow;
  for (int j = tid; j < NV4; j += BLK) {
    float4 v = p4[j];
    kh += (v.x + v.y) + (v.z + v.w);
  }

#if defined(__HIP_DEVICE_COMPILE__)
  asm volatile("s_wait_asynccnt 0" ::: "memory");
#endif
  __syncthreads();

  // ---- e = exp(-z) in place; track zmin/zmax and sum(sigmoid(z)) ----
  float zmn = 3.402823466e38f, zmx = -3.402823466e38f, p0s = 0.f;
  float4* e4 = (float4*)smem;
  for (int j = tid; j < NV4; j += BLK) {
    float4 z = e4[j];
    float4 e;
    e.x = __expf(-z.x); e.y = __expf(-z.y); e.z = __expf(-z.z); e.w = __expf(-z.w);
    zmn = fminf(zmn, fminf(fminf(z.x, z.y), fminf(z.z, z.w)));
    zmx = fmaxf(zmx, fmaxf(fmaxf(z.x, z.y), fmaxf(z.z, z.w)));
    p0s += (frcp(1.f + e.x) + frcp(1.f + e.y)) + (frcp(1.f + e.z) + frcp(1.f + e.w));
    e4[j] = e;
  }

  block_red4(kh, p0s, zmn, zmx, rbuf, tid);  // also fences e writes before first eval

  // ---- Scalar Newton state (uniform across block, recomputed by every thread) ----
  const float Kh = kh;
  const float K = fminf(fmaxf(Kh * 0.25f, 0.f), (float)N_COLS);
  float L = -20.f - zmx;
  float U = 20.f - zmn;
  float lam = logit_clipped(K * (1.f / (float)N_COLS)) -
              logit_clipped(p0s * (1.f / (float)N_COLS));
  lam = fminf(fmaxf(lam, L), U);

  float f, S;
  eval_fS(e4, tid, __expf(-lam), K, rbuf, f, S);

#pragma unroll 1
  for (int it = 0; it < 20; ++it) {
    const bool done = fabsf(f) <= 1e-7f;
    const float newton = lam - f / (S + 1e-18f);
    const bool oob = (newton < L) || (newton > U);
    const float lam_try = (oob || done) ? lam : newton;
    float f_try = f, S_try = S;
    if (lam_try != lam) eval_fS(e4, tid, __expf(-lam_try), K, rbuf, f_try, S_try);
    const bool noimp = (fabsf(f_try) > fabsf(f)) && !oob && !done;
    const float mid = 0.5f * (L + U);
    const float lam_next = (oob || noimp) ? mid : lam_try;
    float f_next = f_try, S_next = S_try;
    if (lam_next != lam_try) eval_fS(e4, tid, __expf(-lam_next), K, rbuf, f_next, S_next);
    if (f_next > 0.f) U = lam_next; else L = lam_next;
    lam = lam_next; f = f_next; S = S_next;
  }
  lam = fminf(fmaxf(lam, L), U);

  // ---- Final loss pass: re-read z and pos from global (L2-resident) ----
  float lsum = 0.f, inter = 0.f, psum = 0.f;
  const float4* z4 = (const float4*)zrow;
  for (int j = tid; j < NV4; j += BLK) {
    float4 z = z4[j];
    float4 pc = p4[j];
#define PROC(ZZ, PP)                                         \
    {                                                        \
      float zs = (ZZ) + lam;                                 \
      float az = fabsf(zs);                                  \
      float mz = fmaxf(zs, 0.f);                             \
      float le = __logf(1.f + __expf(-az));                  \
      lsum += fmaf(4.f, mz + le, -zs * (PP));                \
      float p = frcp(1.f + __expf(-zs));                     \
      inter = fmaf(p, (PP), inter);                          \
      psum += p;                                             \
    }
    PROC(z.x, pc.x) PROC(z.y, pc.y) PROC(z.z, pc.z) PROC(z.w, pc.w)
#undef PROC
  }
  block_red3(lsum, inter, psum, rbuf, tid);
  if (tid == 0) {
    float* w = ws + (size_t)row * 4;
    w[0] = lsum; w[1] = inter; w[2] = psum; w[3] = Kh;
  }
}

__global__ void __launch_bounds__(256, 1)
set_criterion_final_kernel(const float* __restrict__ ws, float* __restrict__ out) {
  __shared__ float r[16];
  const int tid = threadIdx.x;  // 256 threads, one per row
  const float ls = ws[tid * 4 + 0];
  const float it = ws[tid * 4 + 1];
  const float ps = ws[tid * 4 + 2];
  const float kh = ws[tid * 4 + 3];
  float a = ls * (1.f / (512.f * 512.f));
  float d = 1.f - (2.f * it + 1.f) / (fmaf(4.f, ps, kh) + 1.f);
  a = wred_sum(a); d = wred_sum(d);
  const int wid = tid >> 5, lane = tid & 31;
  if (lane == 0) { r[wid] = a; r[8 + wid] = d; }
  __syncthreads();
  if (tid == 0) {
    float sa = 0.f, sd = 0.f;
    for (int i = 0; i < 8; ++i) { sa += r[i]; sd += r[8 + i]; }
    out[0] = sa * (1.f / 256.f);
    out[1] = sd * (1.f / 256.f);
  }
}

extern "C" void kernel_launch(void* const* d_in, const int* in_sizes, int n_in,
                              void* d_out, int out_size, void* d_ws, size_t ws_size,
                              hipStream_t stream) {
  (void)in_sizes; (void)n_in; (void)out_size; (void)ws_size;
  const float* logits = (const float*)d_in[0];
  const float* pos = (const float*)d_in[1];
  float* out = (float*)d_out;
  float* ws = (float*)d_ws;

  const size_t smem_bytes = (size_t)N_COLS * sizeof(float);  // 256 KB / WGP (<=320 KB)
  set_criterion_row_kernel<<<dim3(256), dim3(BLK), smem_bytes, stream>>>(logits, pos, ws);
  set_criterion_final_kernel<<<dim3(1), dim3(256), 0, stream>>>(ws, out);
}